// GATConv_14173392077039
// MI455X (gfx1250) — compile-verified
//
#include <hip/hip_runtime.h>

typedef __attribute__((ext_vector_type(2))) float v2f;
typedef __attribute__((ext_vector_type(8))) float v8f;

#define DIN 256
#define HD  128
#define HH  4
#define DD  32
#define NEG_SLOPE 0.2f

// -------------------------------------------------------------------------
// Kernel 1: ft[N,128] = feat[N,256] @ W[128,256]^T using V_WMMA_F32_16X16X4_F32
// Block = 256 threads = 8 waves. Block covers 16 rows x 128 cols; each wave
// owns one 16x16 output tile and marches K in steps of 4.
// -------------------------------------------------------------------------
__global__ __launch_bounds__(256) void gat_gemm(const float* __restrict__ feat,
                                                const float* __restrict__ W,
                                                float* __restrict__ ft, int N) {
    // 16 rows x 256 K, padded stride 260 => v2f reads hit 64 distinct banks
    __shared__ float As[16][260];
    const int tid = threadIdx.x;
    const int rowBase = blockIdx.x * 16;
    const bool fullTile = (rowBase + 16 <= N);   // block-uniform fast path

    // cooperative stage: 16x256 floats = 1024 float4, 4 per thread
    if (fullTile) {
        #pragma unroll
        for (int i = 0; i < 4; ++i) {
            int idx4 = i * 256 + tid;
            int r    = idx4 >> 6;
            int c4   = idx4 & 63;
            float4 v = reinterpret_cast<const float4*>(feat + (size_t)(rowBase + r) * DIN)[c4];
            As[r][c4 * 4 + 0] = v.x; As[r][c4 * 4 + 1] = v.y;
            As[r][c4 * 4 + 2] = v.z; As[r][c4 * 4 + 3] = v.w;
        }
    } else {
        #pragma unroll
        for (int i = 0; i < 4; ++i) {
            int idx4 = i * 256 + tid;
            int r    = idx4 >> 6;
            int c4   = idx4 & 63;
            float4 v = make_float4(0.f, 0.f, 0.f, 0.f);
            if (rowBase + r < N)
                v = reinterpret_cast<const float4*>(feat + (size_t)(rowBase + r) * DIN)[c4];
            As[r][c4 * 4 + 0] = v.x; As[r][c4 * 4 + 1] = v.y;
            As[r][c4 * 4 + 2] = v.z; As[r][c4 * 4 + 3] = v.w;
        }
    }
    __syncthreads();

    const int wave    = tid >> 5;          // 0..7 -> 16-col tile
    const int lane    = tid & 31;
    const int half    = lane >> 4;         // 0: K0/K1, 1: K2/K3
    const int mrow    = lane & 15;         // A-fragment row
    const int colBase = wave * 16;
    const int ocol    = colBase + (lane & 15); // output channel for B frag
    const int kp      = half * 2;

    v8f c = {0.f, 0.f, 0.f, 0.f, 0.f, 0.f, 0.f, 0.f};
    const float* wrow = W + (size_t)ocol * DIN;

    #pragma unroll 8
    for (int k0 = 0; k0 < DIN; k0 += 4) {
        // A 16x4 f32 fragment: lane<16 holds K0,K1 ; lane>=16 holds K2,K3
        v2f a = *reinterpret_cast<const v2f*>(&As[mrow][k0 + kp]);
        // B 4x16 fragment (B[k][n] = W[n][k]) : contiguous pair from W row
        v2f b = *reinterpret_cast<const v2f*>(wrow + k0 + kp);
        c = __builtin_amdgcn_wmma_f32_16x16x4_f32(
                /*neg_a=*/false, a, /*neg_b=*/false, b,
                /*c_mod=*/(short)0, c, /*reuse_a=*/false, /*reuse_b=*/false);
    }

    // C/D layout: VGPR v -> M = v + 8*half, N = lane&15
    float* outBase = ft + (size_t)(rowBase + 8 * half) * HD + colBase + (lane & 15);
    if (fullTile) {
        #pragma unroll
        for (int v = 0; v < 8; ++v)
            outBase[(size_t)v * HD] = c[v];
    } else {
        #pragma unroll
        for (int v = 0; v < 8; ++v) {
            if (rowBase + v + 8 * half < N)
                outBase[(size_t)v * HD] = c[v];
        }
    }
}

// -------------------------------------------------------------------------
// Kernel 2: el[n,h] = sum_d ft[n,h,d]*attn_l[h,d] ; same for er.
// One block per node, one wave per head (D == warpSize == 32).
// -------------------------------------------------------------------------
__global__ __launch_bounds__(128) void gat_attn_logits(
        const float* __restrict__ ft,
        const float* __restrict__ attn_l, const float* __restrict__ attn_r,
        float* __restrict__ el, float* __restrict__ er) {
    const int n    = blockIdx.x;
    const int h    = threadIdx.x >> 5;
    const int lane = threadIdx.x & 31;
    float v = ft[(size_t)n * HD + h * DD + lane];
    float l = v * attn_l[h * DD + lane];
    float r = v * attn_r[h * DD + lane];
    #pragma unroll
    for (int off = 16; off > 0; off >>= 1) {
        l += __shfl_xor(l, off, 32);
        r += __shfl_xor(r, off, 32);
    }
    if (lane == 0) {
        el[n * HH + h] = l;
        er[n * HH + h] = r;
    }
}

// -------------------------------------------------------------------------
// Kernel 3: out <- bias (broadcast), m <- -inf, denom <- 0
// -------------------------------------------------------------------------
__global__ void gat_init(float* __restrict__ out, const float* __restrict__ bias,
                         float* __restrict__ m, float* __restrict__ denom, int N) {
    size_t i = (size_t)blockIdx.x * blockDim.x + threadIdx.x;
    if (i < (size_t)N * HD) out[i] = bias[i & (HD - 1)];
    if (i < (size_t)N * HH) { m[i] = -3.402823466e38f; denom[i] = 0.f; }
}

__device__ __forceinline__ float leaky(float x) {
    return (x > 0.f) ? x : NEG_SLOPE * x;
}

// -------------------------------------------------------------------------
// Kernel 4: e[E,4] = LeakyReLU(el[src] + er[dst]); segment-max into m[dst].
// -------------------------------------------------------------------------
__global__ void gat_edge_scores(const int* __restrict__ src, const int* __restrict__ dst,
                                const float* __restrict__ el, const float* __restrict__ er,
                                float* __restrict__ esc, float* __restrict__ m, int E) {
    int e = blockIdx.x * blockDim.x + threadIdx.x;
    if (e >= E) return;
    int s = src[e], d = dst[e];
    float4 l = reinterpret_cast<const float4*>(el)[s];
    float4 r = reinterpret_cast<const float4*>(er)[d];
    float4 v;
    v.x = leaky(l.x + r.x); v.y = leaky(l.y + r.y);
    v.z = leaky(l.z + r.z); v.w = leaky(l.w + r.w);
    reinterpret_cast<float4*>(esc)[e] = v;
    atomicMax(&m[d * HH + 0], v.x);
    atomicMax(&m[d * HH + 1], v.y);
    atomicMax(&m[d * HH + 2], v.z);
    atomicMax(&m[d * HH + 3], v.w);
}

// -------------------------------------------------------------------------
// Kernel 5: ae = exp(e - m[dst]); segment-sum into denom[dst]; esc <- ae.
// -------------------------------------------------------------------------
__global__ void gat_edge_exp(const int* __restrict__ dst, float* __restrict__ esc,
                             const float* __restrict__ m, float* __restrict__ denom, int E) {
    int e = blockIdx.x * blockDim.x + threadIdx.x;
    if (e >= E) return;
    int d = dst[e];
    float4 v  = reinterpret_cast<const float4*>(esc)[e];
    float4 mx = reinterpret_cast<const float4*>(m)[d];
    v.x = __expf(v.x - mx.x); v.y = __expf(v.y - mx.y);
    v.z = __expf(v.z - mx.z); v.w = __expf(v.w - mx.w);
    reinterpret_cast<float4*>(esc)[e] = v;
    atomicAdd(&denom[d * HH + 0], v.x);
    atomicAdd(&denom[d * HH + 1], v.y);
    atomicAdd(&denom[d * HH + 2], v.z);
    atomicAdd(&denom[d * HH + 3], v.w);
}

// -------------------------------------------------------------------------
// Kernel 6: out[dst] += ft[src] * (ae/denom[dst]) -- one wave per edge,
// lane = feature within head, h-loop over 4 heads. ft/out are L2-resident
// (51 MB each vs 192 MB L2), so this is L2-atomic-bandwidth bound.
// -------------------------------------------------------------------------
__global__ __launch_bounds__(256) void gat_aggregate(
        const int* __restrict__ src, const int* __restrict__ dst,
        const float* __restrict__ esc, const float* __restrict__ denom,
        const float* __restrict__ ft, float* __restrict__ out, int E) {
    int e = blockIdx.x * 8 + (threadIdx.x >> 5);
    int lane = threadIdx.x & 31;
    if (e >= E) return;
    int s = src[e], d = dst[e];
    const float* fts = ft + (size_t)s * HD;
    __builtin_prefetch(fts + lane, 0, 0);   // global_prefetch_b8: warm the gather row
    float4 ae = reinterpret_cast<const float4*>(esc)[e];
    float4 dn = reinterpret_cast<const float4*>(denom)[d];
    float a0 = ae.x / dn.x, a1 = ae.y / dn.y, a2 = ae.z / dn.z, a3 = ae.w / dn.w;
    float* outd = out + (size_t)d * HD;
    atomicAdd(&outd[0 * DD + lane], fts[0 * DD + lane] * a0);
    atomicAdd(&outd[1 * DD + lane], fts[1 * DD + lane] * a1);
    atomicAdd(&outd[2 * DD + lane], fts[2 * DD + lane] * a2);
    atomicAdd(&outd[3 * DD + lane], fts[3 * DD + lane] * a3);
}

// -------------------------------------------------------------------------
extern "C" void kernel_launch(void* const* d_in, const int* in_sizes, int n_in,
                              void* d_out, int out_size, void* d_ws, size_t ws_size,
                              hipStream_t stream) {
    const float* feat   = (const float*)d_in[0];
    const int*   src    = (const int*)  d_in[1];
    const int*   dst    = (const int*)  d_in[2];
    const float* W      = (const float*)d_in[3];
    const float* attn_l = (const float*)d_in[4];
    const float* attn_r = (const float*)d_in[5];
    const float* bias   = (const float*)d_in[6];

    const int N = in_sizes[0] / DIN;
    const int E = in_sizes[1];
    float* out = (float*)d_out;

    // workspace layout (floats)
    float* ws    = (float*)d_ws;
    float* ft    = ws;                           // N*128
    float* el    = ft    + (size_t)N * HD;       // N*4
    float* er    = el    + (size_t)N * HH;       // N*4
    float* mbuf  = er    + (size_t)N * HH;       // N*4
    float* denom = mbuf  + (size_t)N * HH;       // N*4
    float* esc   = denom + (size_t)N * HH;       // E*4

    gat_gemm<<<(N + 15) / 16, 256, 0, stream>>>(feat, W, ft, N);
    gat_attn_logits<<<N, 128, 0, stream>>>(ft, attn_l, attn_r, el, er);

    size_t tot = (size_t)N * HD;
    gat_init<<<(unsigned)((tot + 255) / 256), 256, 0, stream>>>(out, bias, mbuf, denom, N);

    gat_edge_scores<<<(E + 255) / 256, 256, 0, stream>>>(src, dst, el, er, esc, mbuf, E);
    gat_edge_exp<<<(E + 255) / 256, 256, 0, stream>>>(dst, esc, mbuf, denom, E);
    gat_aggregate<<<(E + 7) / 8, 256, 0, stream>>>(src, dst, esc, denom, ft, out, E);
}